// DistSageConv_10230612099179
// MI455X (gfx1250) — compile-verified
//
#include <hip/hip_runtime.h>
#include <hip/hip_bf16.h>

// DistSageConv on MI455X (gfx1250):
//   result = segsum(x[src]) @ W1^T + cnt*b1  +  x[:n_dst] @ W2^T + b2
// CSR-build -> no-atomic aggregate -> pack B -> fused f32 WMMA GEMM
// (M=40000, N=128, K=512 with A = [agg | x]).

typedef __attribute__((ext_vector_type(2))) float v2f;   // WMMA A/B frag (16x4 f32)
typedef __attribute__((ext_vector_type(8))) float v8f;   // WMMA C/D frag (16x16 f32)
typedef __attribute__((ext_vector_type(4))) float f4;

#define IN_F   256
#define OUT_F  128
#define KTOT   512          // 2 * IN_F  (concatenated K)
#define BM     128          // M rows per workgroup
#define KC     64           // K per LDS stage
#define NSTAGE (KTOT / KC)  // 8
#define ASTRIDE 68          // 64 + 4 pad  -> conflict-free A frag reads
#define BSTRIDE 288         // 2*128 + 32 pad -> half-waves hit disjoint bank sets

// ---------------- Kernel 1: CSR rowptr from sorted dst_ids (no atomics) -------------
__global__ void k_rowptr(const int* __restrict__ dst, int n_edges,
                         int* __restrict__ rowptr, int n_dst) {
  int i = blockIdx.x * blockDim.x + threadIdx.x;
  if (i > n_edges) return;
  int cur  = (i < n_edges) ? dst[i] : n_dst;
  int prev = (i == 0) ? -1 : dst[i - 1];
  if (cur > n_dst) cur = n_dst;
  for (int d = prev + 1; d <= cur; ++d) rowptr[d] = i;   // rowptr[d] = first edge with dst >= d
}

// ---------------- Kernel 2: per-dst feature aggregation (atomic-free) ---------------
__global__ void k_aggregate(const float* __restrict__ x,
                            const int* __restrict__ src_ids,
                            const int* __restrict__ rowptr,
                            float* __restrict__ agg,
                            int* __restrict__ cnt) {
  int d = blockIdx.x;
  int t = threadIdx.x;                 // feature column 0..255
  int s = rowptr[d], e = rowptr[d + 1];
  float acc = 0.f;
  for (int j = s; j < e; ++j) {
    int src = src_ids[j];
    if (j + 1 < e) {
      // CDNA5 global_prefetch_b8: warm the next gathered row
      __builtin_prefetch(x + (size_t)src_ids[j + 1] * IN_F + t, 0, 0);
    }
    acc += x[(size_t)src * IN_F + t];
  }
  agg[(size_t)d * IN_F + t] = acc;
  if (t == 0) cnt[d] = e - s;
}

// ---------------- Kernel 3: pack B = [W1 | W2]^T, K-pair interleaved ----------------
// Bpack flat index: (k>>1)*256 + 2*n + (k&1)  => B fragment loads become ds_load_b64.
__global__ void k_packB(const float* __restrict__ W1, const float* __restrict__ W2,
                        float* __restrict__ Bpack) {
  int t = blockIdx.x * blockDim.x + threadIdx.x;
  if (t >= KTOT * OUT_F) return;
  int k = t >> 7;          // 0..511
  int n = t & 127;         // 0..127
  float v = (k < IN_F) ? W1[n * IN_F + k] : W2[n * IN_F + (k - IN_F)];
  Bpack[(k >> 1) * (2 * OUT_F) + 2 * n + (k & 1)] = v;
}

// ---------------- Kernel 4: fused WMMA f32 GEMM + bias epilogue ---------------------
__global__ void __launch_bounds__(256)
k_gemm(const float* __restrict__ agg, const float* __restrict__ x,
       const float* __restrict__ Bpack, const int* __restrict__ cnt,
       const float* __restrict__ b1, const float* __restrict__ b2,
       float* __restrict__ out, int n_dst) {
  __shared__ float As[2][BM * ASTRIDE];          // 2 * 128*68*4  = 69.6 KB
  __shared__ float Bs[2][(KC / 2) * BSTRIDE];    // 2 * 32*288*4  = 73.7 KB

  const int m0 = blockIdx.x * BM;
  const int t  = threadIdx.x;

  // staging maps: A -> thread loads 32 floats of one row half; B -> 32 floats of one k-pair row
  const int ar = t >> 1, aq = (t & 1) * 32;
  const int bkp = t >> 3, bj = (t & 7) * 32;

  f4 aReg[8], bReg[8];

  auto gloadA = [&](int s) {
    const float* src = (s < 4) ? agg : x;        // K 0..255 = agg, 256..511 = x
    int k0 = (s & 3) * KC;
    int row = m0 + ar; if (row >= n_dst) row = n_dst - 1;
    const float* p = src + (size_t)row * IN_F + k0 + aq;
#pragma unroll
    for (int i = 0; i < 8; ++i) aReg[i] = *(const f4*)(p + 4 * i);
  };
  auto gloadB = [&](int s) {
    const float* p = Bpack + (size_t)(s * (KC / 2) + bkp) * (2 * OUT_F) + bj;
#pragma unroll
    for (int i = 0; i < 8; ++i) bReg[i] = *(const f4*)(p + 4 * i);
  };
  auto lstore = [&](int buf) {
    float* ap = &As[buf][ar * ASTRIDE + aq];
#pragma unroll
    for (int i = 0; i < 8; ++i) *(f4*)(ap + 4 * i) = aReg[i];
    float* bp = &Bs[buf][bkp * BSTRIDE + bj];
#pragma unroll
    for (int i = 0; i < 8; ++i) *(f4*)(bp + 4 * i) = bReg[i];
  };

  // wave layout: 8 waves = 4 (M subtiles of 32 rows) x 2 (N halves);
  // each wave owns a 32x64 output strip = 8 accumulators -> 8 WMMAs per 6 LDS frag loads.
  const int lane = t & 31;
  const int wave = t >> 5;
  const int wm = wave & 3;          // 32-row M group
  const int wn = wave >> 2;         // 64-col N half
  const int row = lane & 15;
  const int hi  = lane >> 4;        // half-wave select (K offset 0/2)

  v8f c[8] = {v8f{}, v8f{}, v8f{}, v8f{}, v8f{}, v8f{}, v8f{}, v8f{}};

  auto compute = [&](int buf) {
    const float* abase = &As[buf][(32 * wm + row) * ASTRIDE + 2 * hi];
    const float* bbase = &Bs[buf][hi * BSTRIDE + 2 * (64 * wn + row)];
#pragma unroll
    for (int kk = 0; kk < KC; kk += 4) {
      v2f a0 = *(const v2f*)(abase + kk);                     // rows 32wm..+15
      v2f a1 = *(const v2f*)(abase + 16 * ASTRIDE + kk);      // rows 32wm+16..+31
      const float* bb = bbase + (kk >> 1) * BSTRIDE;
      v2f b0 = *(const v2f*)(bb + 0);                         // 4 N-subtiles
      v2f b1v = *(const v2f*)(bb + 32);
      v2f b2v = *(const v2f*)(bb + 64);
      v2f b3v = *(const v2f*)(bb + 96);
      c[0] = __builtin_amdgcn_wmma_f32_16x16x4_f32(false, a0, false, b0,  (short)0, c[0], false, false);
      c[1] = __builtin_amdgcn_wmma_f32_16x16x4_f32(false, a0, false, b1v, (short)0, c[1], false, false);
      c[2] = __builtin_amdgcn_wmma_f32_16x16x4_f32(false, a0, false, b2v, (short)0, c[2], false, false);
      c[3] = __builtin_amdgcn_wmma_f32_16x16x4_f32(false, a0, false, b3v, (short)0, c[3], false, false);
      c[4] = __builtin_amdgcn_wmma_f32_16x16x4_f32(false, a1, false, b0,  (short)0, c[4], false, false);
      c[5] = __builtin_amdgcn_wmma_f32_16x16x4_f32(false, a1, false, b1v, (short)0, c[5], false, false);
      c[6] = __builtin_amdgcn_wmma_f32_16x16x4_f32(false, a1, false, b2v, (short)0, c[6], false, false);
      c[7] = __builtin_amdgcn_wmma_f32_16x16x4_f32(false, a1, false, b3v, (short)0, c[7], false, false);
    }
  };

  // ---- software-pipelined K loop (double-buffered LDS) ----
  gloadA(0); gloadB(0);
  lstore(0);
  __syncthreads();
#pragma unroll 1
  for (int s = 0; s < NSTAGE; ++s) {
    if (s + 1 < NSTAGE) { gloadA(s + 1); gloadB(s + 1); }   // overlap global latency
    compute(s & 1);
    if (s + 1 < NSTAGE) {
      __syncthreads();
      lstore((s + 1) & 1);
      __syncthreads();
    }
  }

  // ---- epilogue: + cnt*b1 + b2 ----
#pragma unroll
  for (int j = 0; j < 4; ++j) {
    int n = 64 * wn + 16 * j + row;
    float bb1 = b1[n], bb2 = b2[n];
#pragma unroll
    for (int i = 0; i < 2; ++i) {
#pragma unroll
      for (int v = 0; v < 8; ++v) {
        int r = m0 + 32 * wm + 16 * i + v + 8 * hi;          // C/D layout: VGPR v -> M=v | v+8
        if (r < n_dst)
          out[(size_t)r * OUT_F + n] = c[i * 4 + j][v] + (float)cnt[r] * bb1 + bb2;
      }
    }
  }
}

// ------------------------------------------------------------------------------------
extern "C" void kernel_launch(void* const* d_in, const int* in_sizes, int n_in,
                              void* d_out, int out_size, void* d_ws, size_t ws_size,
                              hipStream_t stream) {
  const float* x   = (const float*)d_in[0];
  const float* W1  = (const float*)d_in[1];
  const float* b1  = (const float*)d_in[2];
  const float* W2  = (const float*)d_in[3];
  const float* b2  = (const float*)d_in[4];
  const int* src_ids = (const int*)d_in[5];
  const int* dst_ids = (const int*)d_in[6];
  const int n_edges = in_sizes[5];
  const int n_dst   = out_size / OUT_F;      // 40000
  float* out = (float*)d_out;

  // workspace carve-out (~41.5 MB): rowptr | cnt | Bpack | agg
  char* ws = (char*)d_ws;
  size_t off = 0;
  auto carve = [&](size_t bytes) -> void* {
    void* p = ws + off;
    off += (bytes + 255) & ~(size_t)255;
    return p;
  };
  int*   rowptr = (int*)  carve(sizeof(int)   * (size_t)(n_dst + 1));
  int*   cnt    = (int*)  carve(sizeof(int)   * (size_t)n_dst);
  float* Bpack  = (float*)carve(sizeof(float) * (size_t)KTOT * OUT_F);
  float* agg    = (float*)carve(sizeof(float) * (size_t)n_dst * IN_F);
  (void)ws_size; (void)n_in;

  k_rowptr<<<(n_edges + 1 + 255) / 256, 256, 0, stream>>>(dst_ids, n_edges, rowptr, n_dst);
  k_aggregate<<<n_dst, IN_F, 0, stream>>>(x, src_ids, rowptr, agg, cnt);
  k_packB<<<(KTOT * OUT_F + 255) / 256, 256, 0, stream>>>(W1, W2, Bpack);
  k_gemm<<<(n_dst + BM - 1) / BM, 256, 0, stream>>>(agg, x, Bpack, cnt, b1, b2, out, n_dst);
}